// CBF_31937376813309
// MI455X (gfx1250) — compile-verified
//
#include <hip/hip_runtime.h>
#include <hip/hip_bf16.h>

#define NAGENT 4096
#define TOPK   12
#define EPSF   1e-4f

typedef __attribute__((ext_vector_type(2))) float v2f;
typedef __attribute__((ext_vector_type(8))) float v8f;

// ---------------------------------------------------------------------------
// Phase 1: per-agent top-K (K=12) nearest neighbors by planar distance.
// One thread per row i; states' xy cached in LDS (32KB) per 256-thread block.
// 12-slot list sorted ascending by (d^2, j) — monotonic with the reference
// key sqrt(d^2 + 2*eps) and matching lax.top_k tie-breaking.
// ---------------------------------------------------------------------------
__global__ __launch_bounds__(256)
void CBF_topk_kernel(const float* __restrict__ states, int* __restrict__ topk) {
    __shared__ float sx[NAGENT];
    __shared__ float sy[NAGENT];
    const int tid = threadIdx.x;
    for (int j = tid; j < NAGENT; j += 256) {
        sx[j] = states[4 * j + 0];
        sy[j] = states[4 * j + 1];
    }
    __syncthreads();

    const int i = blockIdx.x * 256 + tid;
    const float xi = sx[i];
    const float yi = sy[i];

    float bd[TOPK];
    int   bi[TOPK];
#pragma unroll
    for (int k = 0; k < TOPK; ++k) { bd[k] = 3.4e38f; bi[k] = 0x7fffffff; }

    for (int j = 0; j < NAGENT; ++j) {
        const float dx = xi - sx[j];
        const float dy = yi - sy[j];
        const float d  = dx * dx + dy * dy;
        if (d < bd[TOPK - 1] || (d == bd[TOPK - 1] && j < bi[TOPK - 1])) {
            int p = TOPK - 1;
            while (p > 0 && (d < bd[p - 1] || (d == bd[p - 1] && j < bi[p - 1]))) {
                bd[p] = bd[p - 1];
                bi[p] = bi[p - 1];
                --p;
            }
            bd[p] = d;
            bi[p] = j;
        }
    }
#pragma unroll
    for (int k = 0; k < TOPK; ++k) topk[i * TOPK + k] = bi[k];
}

// ---------------------------------------------------------------------------
// One fp32 WMMA layer step: [16 x KIN] (LDS, stride 128) x [KW x NOUT] (global,
// zero-padded at compile time to KIN rows) + bias, optional ReLU -> LDS.
// V_WMMA_F32_16X16X4_F32 tiles: M=16, N-tiles of 16, K-chunks of 4.
// A frag (16x4 f32): lanes 0-15 hold K={0,1}, lanes 16-31 hold K={2,3}.
// B frag (4x16 f32): VGPR0 = rows {0,2}, VGPR1 = rows {1,3} (lane-half select).
// C/D (16x16 f32):   vgpr v, lane l -> row v + 8*(l>=16), col l%16.
// All loops fully unrolled: KW guards fold to constants (no exec branches),
// A fragments hoisted to registers once per layer (single LDS read per chunk).
// ---------------------------------------------------------------------------
template <int KIN, int KW, int NOUT, bool RELU>
__device__ __forceinline__ void mlp_layer(const float* __restrict__ W,
                                          const float* __restrict__ bias,
                                          const float* bufIn, float* bufOut,
                                          int lane) {
    const int row  = lane & 15;
    const int half = lane >> 4;

    v2f afrag[KIN / 4];
#pragma unroll
    for (int kc = 0; kc < KIN / 4; ++kc) {
        afrag[kc] = *(const v2f*)(bufIn + row * 128 + kc * 4 + 2 * half);
    }

#pragma unroll
    for (int nt = 0; nt < NOUT / 16; ++nt) {
        const int   n  = nt * 16 + row;
        const float bn = bias[n];
        v8f c;
#pragma unroll
        for (int v = 0; v < 8; ++v) c[v] = bn;  // bias pre-loaded into C

#pragma unroll
        for (int kc = 0; kc < KIN / 4; ++kc) {
            const int cb = kc * 4 + 2 * half;
            v2f b;
            // cb < KW is compile-time decidable after full unroll for the
            // half=0/half=1 split only in the padded tail chunk; out-of-range
            // rows use a clamped (in-bounds) address and a select to 0.
            const int r0 = (kc * 4 + 2 * half < KW) ? (kc * 4 + 2 * half) : 0;
            const int r1 = (kc * 4 + 2 * half + 1 < KW) ? (kc * 4 + 2 * half + 1) : 0;
            const float w0 = W[r0 * NOUT + n];
            const float w1 = W[r1 * NOUT + n];
            b.x = (cb     < KW) ? w0 : 0.0f;
            b.y = (cb + 1 < KW) ? w1 : 0.0f;
            c = __builtin_amdgcn_wmma_f32_16x16x4_f32(
                    false, afrag[kc], false, b, (short)0, c, false, false);
        }
#pragma unroll
        for (int v = 0; v < 8; ++v) {
            float x = c[v];
            if (RELU) x = fmaxf(x, 0.0f);
            bufOut[(v + 8 * half) * 128 + nt * 16 + row] = x;
        }
    }
}

// ---------------------------------------------------------------------------
// Phase 2: per-edge MLP 6->64->128->64->1 on WMMA fp32 pipes.
// One wave per 16-edge tile (49152 edges -> 3072 waves). Activations ping-pong
// between two 16x128 LDS buffers; EXEC is all-ones at every WMMA.
// ---------------------------------------------------------------------------
__global__ __launch_bounds__(32)
void CBF_mlp_kernel(const float4* __restrict__ states,
                    const int* __restrict__ topk,
                    const float* __restrict__ W1, const float* __restrict__ b1,
                    const float* __restrict__ W2, const float* __restrict__ b2,
                    const float* __restrict__ W3, const float* __restrict__ b3,
                    const float* __restrict__ W4, const float* __restrict__ b4,
                    float* __restrict__ out) {
    __shared__ __align__(16) float act0[16 * 128];
    __shared__ __align__(16) float act1[16 * 128];
    __shared__ float smask[16];

    const int lane = threadIdx.x;
    const int e0   = blockIdx.x * 16;

    // ---- feature build: feat = [dx,dy,dz,dw, eye, dnorm-0.1], pad to 8 ----
    if (lane < 16) {
        const int    e  = e0 + lane;
        const int    i  = e / TOPK;
        const int    j  = topk[e];
        const float4 si = states[i];
        const float4 sj = states[j];
        const float dx = si.x - sj.x;
        const float dy = si.y - sj.y;
        const float dz = si.z - sj.z;
        const float dw = si.w - sj.w;
        const float dist = sqrtf(dx * dx + dy * dy + 2.0f * EPSF);
        float* f = act0 + lane * 128;
        f[0] = dx;
        f[1] = dy;
        f[2] = dz;
        f[3] = dw;
        f[4] = (i == j) ? 1.0f : 0.0f;
        f[5] = dist - 0.1f;
        f[6] = 0.0f;   // K-pad
        f[7] = 0.0f;   // K-pad
        smask[lane] = (dist <= 1.0f) ? 1.0f : 0.0f;
    }
    __syncthreads();

    mlp_layer<8,   6,   64,  true>(W1, b1, act0, act1, lane);
    __syncthreads();
    mlp_layer<64,  64,  128, true>(W2, b2, act1, act0, lane);
    __syncthreads();
    mlp_layer<128, 128, 64,  true>(W3, b3, act0, act1, lane);
    __syncthreads();

    // ---- head: 64 -> 1, N padded to 16 (only column 0 is live) ----
    // Unconditional W4 loads (cb+1 <= 63, always in-bounds) + lane select;
    // no exec-mask branches around the loads.
    const int  row  = lane & 15;
    const int  half = lane >> 4;
    const bool col0 = (row == 0);
    const float b4v = b4[0];
    v8f c;
#pragma unroll
    for (int v = 0; v < 8; ++v) c[v] = col0 ? b4v : 0.0f;

#pragma unroll
    for (int kc = 0; kc < 16; ++kc) {
        const int cb = kc * 4 + 2 * half;
        v2f a = *(const v2f*)(act1 + row * 128 + cb);
        v2f w = *(const v2f*)(W4 + cb);   // global_load_b64, in-bounds for all lanes
        v2f b;
        b.x = col0 ? w.x : 0.0f;
        b.y = col0 ? w.y : 0.0f;
        c = __builtin_amdgcn_wmma_f32_16x16x4_f32(
                false, a, false, b, (short)0, c, false, false);
    }

    // column 0 lives in lane 0 (rows 0-7) and lane 16 (rows 8-15)
    if (col0) {
#pragma unroll
        for (int v = 0; v < 8; ++v) {
            const int m = v + 8 * half;
            out[e0 + m] = c[v] * smask[m];
        }
    }
}

extern "C" void kernel_launch(void* const* d_in, const int* in_sizes, int n_in,
                              void* d_out, int out_size, void* d_ws, size_t ws_size,
                              hipStream_t stream) {
    const float* states = (const float*)d_in[0];
    const float* W1     = (const float*)d_in[1];
    const float* b1     = (const float*)d_in[2];
    const float* W2     = (const float*)d_in[3];
    const float* b2     = (const float*)d_in[4];
    const float* W3     = (const float*)d_in[5];
    const float* b3     = (const float*)d_in[6];
    const float* W4     = (const float*)d_in[7];
    const float* b4     = (const float*)d_in[8];
    float*       out    = (float*)d_out;

    int* topk = (int*)d_ws;  // NAGENT * TOPK ints = 192 KiB of scratch

    CBF_topk_kernel<<<NAGENT / 256, 256, 0, stream>>>(states, topk);

    const int n_tiles = (NAGENT * TOPK) / 16;  // 3072 tiles of 16 edges
    CBF_mlp_kernel<<<n_tiles, 32, 0, stream>>>(
        (const float4*)states, topk,
        W1, b1, W2, b2, W3, b3, W4, b4, out);
}